// LinearEncoder_64785286693394
// MI455X (gfx1250) — compile-verified
//
#include <hip/hip_runtime.h>
#include <hip/hip_bf16.h>
#include <math.h>

typedef __attribute__((ext_vector_type(2))) float v2f;
typedef __attribute__((ext_vector_type(8))) float v8f;

#define D 128   // D_IN == D_OUT == 128

// ---------------- degree kernels ----------------

__global__ __launch_bounds__(256) void deg_init_kernel(float* __restrict__ deg, int n) {
    int i = blockIdx.x * blockDim.x + threadIdx.x;
    if (i < n) deg[i] = 1.0f;   // self-loop
}

__global__ __launch_bounds__(256) void deg_count_kernel(const int* __restrict__ dst,
                                                        float* __restrict__ deg, int e) {
    int i = blockIdx.x * blockDim.x + threadIdx.x;
    if (i < e) atomicAdd(&deg[dst[i]], 1.0f);
}

__global__ __launch_bounds__(256) void deg_to_dis_kernel(float* __restrict__ deg, int n) {
    int i = blockIdx.x * blockDim.x + threadIdx.x;
    if (i < n) deg[i] = rsqrtf(deg[i]);   // deg >= 1 always
}

// ---------------- h = x @ W via fp32 WMMA 16x16x4 ----------------
// 4 waves per 128-thread block; wave w computes rows [ (4*bid+w)*16 , +16 )
// across all 128 output columns (8 tiles of 16). K loop: 32 steps of 4.

__global__ __launch_bounds__(128) void gemm_wmma_kernel(const float* __restrict__ x,
                                                        const float* __restrict__ W,
                                                        float* __restrict__ h, int n) {
    __shared__ float sW[D * D];   // 64 KB, fits CDNA5 WGP LDS

    int tid = threadIdx.x;
    // cooperative load of W into LDS (float4, 32 iters/thread)
    {
        const float4* Wv  = (const float4*)W;
        float4*       sWv = (float4*)sW;
        #pragma unroll
        for (int i = tid; i < (D * D) / 4; i += 128) sWv[i] = Wv[i];
    }
    __syncthreads();

    const int wave = tid >> 5;
    const int lane = tid & 31;
    const int row0 = (blockIdx.x * 4 + wave) * 16;
    if (row0 >= n) return;                    // wave-uniform exit
    const bool fullTile = (row0 + 16 <= n);   // wave-uniform (always true when n % 16 == 0)

    const int m     = lane & 15;           // row-in-tile (A) / col-in-tile (B,C)
    const int khalf = (lane >> 4) * 2;     // lanes 0-15: K=k0,k0+1 ; lanes 16-31: K=k0+2,k0+3
    const int row   = row0 + m;

    v8f acc[8] = {};                       // 8 column tiles of 16

    if (fullTile) {
        const float* xrow = x + (size_t)row * D;
        for (int k0 = 0; k0 < D; k0 += 4) {
            v2f a;
            a.x = xrow[k0 + khalf];
            a.y = xrow[k0 + khalf + 1];
            #pragma unroll
            for (int t = 0; t < 8; ++t) {
                const int n0 = t * 16;
                v2f b;
                b.x = sW[(k0 + khalf)     * D + n0 + m];
                b.y = sW[(k0 + khalf + 1) * D + n0 + m];
                acc[t] = __builtin_amdgcn_wmma_f32_16x16x4_f32(
                    false, a, false, b, (short)0, acc[t], false, false);
            }
        }
        // C layout: VGPR r -> row r (lanes 0-15) / row r+8 (lanes 16-31), col = lane&15
        float* hp = h + (size_t)(row0 + ((lane >> 4) * 8)) * D + m;
        #pragma unroll
        for (int t = 0; t < 8; ++t) {
            #pragma unroll
            for (int r = 0; r < 8; ++r) {
                hp[(size_t)r * D + t * 16] = acc[t][r];
            }
        }
    } else {
        const bool rowOk = (row < n);
        const float* xrow = x + (size_t)(rowOk ? row : 0) * D;
        for (int k0 = 0; k0 < D; k0 += 4) {
            v2f a;
            float a0 = xrow[k0 + khalf];
            float a1 = xrow[k0 + khalf + 1];
            a.x = rowOk ? a0 : 0.0f;
            a.y = rowOk ? a1 : 0.0f;
            #pragma unroll
            for (int t = 0; t < 8; ++t) {
                const int n0 = t * 16;
                v2f b;
                b.x = sW[(k0 + khalf)     * D + n0 + m];
                b.y = sW[(k0 + khalf + 1) * D + n0 + m];
                acc[t] = __builtin_amdgcn_wmma_f32_16x16x4_f32(
                    false, a, false, b, (short)0, acc[t], false, false);
            }
        }
        const int rbase = row0 + ((lane >> 4) * 8);
        #pragma unroll
        for (int t = 0; t < 8; ++t) {
            #pragma unroll
            for (int r = 0; r < 8; ++r) {
                const int rr = rbase + r;
                if (rr < n) h[(size_t)rr * D + t * 16 + m] = acc[t][r];
            }
        }
    }
}

// ---------------- out = b + h * (1/deg)  (self-loop term), float4 ----------------

__global__ __launch_bounds__(256) void out_init_kernel(const float4* __restrict__ h4,
                                                       const float* __restrict__ dis,
                                                       const float4* __restrict__ b4,
                                                       float4* __restrict__ out4, int n) {
    int idx = blockIdx.x * blockDim.x + threadIdx.x;   // over n * 32 float4s
    if (idx < n * (D / 4)) {
        int i = idx >> 5;            // node
        int j = idx & ((D / 4) - 1); // float4 index within row
        float d  = dis[i];
        float nm = d * d;            // self-loop norm = 1/deg
        float4 hv = h4[idx];
        float4 bv = b4[j];
        float4 o;
        o.x = bv.x + hv.x * nm;
        o.y = bv.y + hv.y * nm;
        o.z = bv.z + hv.z * nm;
        o.w = bv.w + hv.w * nm;
        out4[idx] = o;
    }
}

// ---------------- edge scatter: wave per edge, float4 + 4 atomics/lane ----------------

__global__ __launch_bounds__(256) void scatter_kernel(const float* __restrict__ h,
                                                      const float* __restrict__ dis,
                                                      const int* __restrict__ src,
                                                      const int* __restrict__ dst,
                                                      float* __restrict__ out, int e) {
    int edge = blockIdx.x * (blockDim.x >> 5) + (threadIdx.x >> 5);
    if (edge >= e) return;
    int lane = threadIdx.x & 31;

    int s = src[edge];           // wave-uniform
    int d = dst[edge];
    float norm = dis[s] * dis[d];

    const float4* hv = (const float4*)(h + (size_t)s * D);
    float4 v = hv[lane];         // 32 lanes x 16B = coalesced 512B row

    float* o = out + (size_t)d * D + lane * 4;
    atomicAdd(o + 0, v.x * norm);
    atomicAdd(o + 1, v.y * norm);
    atomicAdd(o + 2, v.z * norm);
    atomicAdd(o + 3, v.w * norm);
}

// ---------------- host entry ----------------

extern "C" void kernel_launch(void* const* d_in, const int* in_sizes, int n_in,
                              void* d_out, int out_size, void* d_ws, size_t ws_size,
                              hipStream_t stream) {
    const float* x  = (const float*)d_in[0];
    const int*   ei = (const int*)d_in[1];     // [2, E] flat: row 0 = src, row 1 = dst
    const float* W  = (const float*)d_in[2];
    const float* b  = (const float*)d_in[3];
    float* out = (float*)d_out;

    const int n = in_sizes[0] / D;     // 50000
    const int e = in_sizes[1] / 2;     // 600000
    const int* src = ei;
    const int* dst = ei + e;

    // workspace: h [n*D floats] then deg/dis [n floats]
    float* h   = (float*)d_ws;
    float* dis = h + (size_t)n * D;

    // 1) degrees (float atomics, exact for counts < 2^24)
    deg_init_kernel<<<(n + 255) / 256, 256, 0, stream>>>(dis, n);
    deg_count_kernel<<<(e + 255) / 256, 256, 0, stream>>>(dst, dis, e);
    deg_to_dis_kernel<<<(n + 255) / 256, 256, 0, stream>>>(dis, n);

    // 2) h = x @ W  (fp32 WMMA). 16-row tiles, 4 waves/block.
    const int rowTiles = (n + 15) / 16;
    gemm_wmma_kernel<<<(rowTiles + 3) / 4, 128, 0, stream>>>(x, W, h, n);

    // 3) out = b + h/deg  (also fully initializes poisoned d_out)
    const int q = n * (D / 4);
    out_init_kernel<<<(q + 255) / 256, 256, 0, stream>>>(
        (const float4*)h, dis, (const float4*)b, (float4*)out, n);

    // 4) edge scatter-add (wave per edge)
    const int edgesPerBlock = 256 / 32;
    scatter_kernel<<<(e + edgesPerBlock - 1) / edgesPerBlock, 256, 0, stream>>>(
        h, dis, src, dst, out, e);
}